// TreeNet_26388279066669
// MI455X (gfx1250) — compile-verified
//
#include <hip/hip_runtime.h>
#include <hip/hip_bf16.h>
#include <math.h>

// ---------------- problem constants ----------------
#define HID   150
#define HP    160          // padded hidden (10 x 16 tiles)
#define EMB   300          // multiple of 4, no K padding needed
#define NNODE 8191
#define NPADR 8192
#define NLEAF 4096
#define LEAF0 4095         // leaf_start = 2^12 - 1
#define KPAD  160          // padded K for internal GEMMs

typedef __attribute__((ext_vector_type(2))) float v2f;
typedef __attribute__((ext_vector_type(8))) float v8f;

__device__ __forceinline__ v8f wmma4(v2f a, v2f b, v8f c) {
  // D = A(16x4,f32) x B(4x16,f32) + C(16x16,f32)  -> v_wmma_f32_16x16x4_f32
  return __builtin_amdgcn_wmma_f32_16x16x4_f32(
      /*neg_a=*/false, a, /*neg_b=*/false, b,
      /*c_mod=*/(short)0, c, /*reuse_a=*/false, /*reuse_b=*/false);
}

__device__ __forceinline__ float sigf(float x) { return 1.0f / (1.0f + expf(-x)); }

// ---------------- weight pack: W(rows x cols, row-major) -> PK[(k/2)*npad + n] = {W[n][k], W[n][k+1]}, zero padded
__global__ void pack_weight(const float* __restrict__ W, v2f* __restrict__ PK,
                            int rows, int cols, int kpairs, int npad) {
  int idx = blockIdx.x * blockDim.x + threadIdx.x;
  int total = kpairs * npad;
  if (idx >= total) return;
  int kh = idx / npad;
  int n  = idx - kh * npad;
  int k0 = kh * 2;
  v2f v;
  v.x = (n < rows && k0     < cols) ? W[n * cols + k0    ] : 0.0f;
  v.y = (n < rows && k0 + 1 < cols) ? W[n * cols + k0 + 1] : 0.0f;
  PK[idx] = v;
}

// ---------------- leaf kernel: emb @ {Wi,Wo,Wu}^T -> gates -> h,c ----------------
__global__ __launch_bounds__(32) void leaf_kernel(
    const float* __restrict__ emb, const int* __restrict__ words,
    const v2f* __restrict__ WiPK, const v2f* __restrict__ WoPK, const v2f* __restrict__ WuPK,
    const float* __restrict__ bi, const float* __restrict__ bo, const float* __restrict__ bu,
    float* __restrict__ H, float* __restrict__ C) {
  const int lane = threadIdx.x;
  const int l15  = lane & 15;
  const int half = lane >> 4;
  const int koff = half << 1;
  const int tm = blockIdx.x, tn = blockIdx.y;
  const int row = tm * 16 + l15;            // A-fragment row (same for both half-waves)
  const int col = tn * 16 + l15;            // B/D column

  const long  word = (long)words[LEAF0 + row];
  const float* arow = emb + word * (long)EMB;

  v8f aI = {}, aO = {}, aU = {};
  for (int k = 0; k < EMB; k += 4) {
    v2f a  = *(const v2f*)(arow + k + koff);
    int kh = ((k + koff) >> 1) * HP + col;
    aI = wmma4(a, WiPK[kh], aI);
    aO = wmma4(a, WoPK[kh], aO);
    aU = wmma4(a, WuPK[kh], aU);
  }

  const int  cc  = min(col, HID - 1);
  const bool pad = (col >= HID);
  const float bic = bi[cc], boc = bo[cc], buc = bu[cc];
#pragma unroll
  for (int r = 0; r < 8; ++r) {
    int rowd = tm * 16 + r + (half << 3);   // D row for this accumulator slot
    float ig = sigf(aI[r] + bic);
    float og = sigf(aO[r] + boc);
    float uv = tanhf(aU[r] + buc);
    float c  = ig * uv;
    float h  = og * tanhf(c);
    size_t off = (size_t)(LEAF0 + rowd) * HP + col;
    H[off] = pad ? 0.0f : h;
    C[off] = pad ? 0.0f : c;
  }
}

// ---------------- internal level kernel: 10 GEMMs + LSTM cell ----------------
__global__ __launch_bounds__(32) void internal_kernel(
    const int* __restrict__ lchs, const int* __restrict__ rchs,
    const v2f* __restrict__ PKi0, const v2f* __restrict__ PKi1,
    const v2f* __restrict__ PKf00, const v2f* __restrict__ PKf01,
    const v2f* __restrict__ PKf10, const v2f* __restrict__ PKf11,
    const v2f* __restrict__ PKo0, const v2f* __restrict__ PKo1,
    const v2f* __restrict__ PKu0, const v2f* __restrict__ PKu1,
    const float* __restrict__ bbi, const float* __restrict__ bbf,
    const float* __restrict__ bbo, const float* __restrict__ bbu,
    float* __restrict__ H, float* __restrict__ C, int s, int W) {
  const int lane = threadIdx.x;
  const int l15  = lane & 15;
  const int half = lane >> 4;
  const int koff = half << 1;
  const int tm = blockIdx.x, tn = blockIdx.y;
  const int row  = tm * 16 + l15;
  const int rowc = min(row, W - 1);         // clamp: keep EXEC all-ones for WMMA
  const int node = s + rowc;
  const long li = lchs[node], ri = rchs[node];
  const float* lrow = H + (size_t)li * HP;
  const float* rrow = H + (size_t)ri * HP;
  const int col = tn * 16 + l15;

  v8f aI = {}, aFl = {}, aFr = {}, aO = {}, aU = {};
  for (int k = 0; k < KPAD; k += 4) {
    v2f al = *(const v2f*)(lrow + k + koff);
    v2f ar = *(const v2f*)(rrow + k + koff);
    int kh = ((k + koff) >> 1) * HP + col;
    aI  = wmma4(al, PKi0 [kh], aI );  aI  = wmma4(ar, PKi1 [kh], aI );
    aFl = wmma4(al, PKf00[kh], aFl);  aFl = wmma4(ar, PKf01[kh], aFl);
    aFr = wmma4(al, PKf10[kh], aFr);  aFr = wmma4(ar, PKf11[kh], aFr);
    aO  = wmma4(al, PKo0 [kh], aO );  aO  = wmma4(ar, PKo1 [kh], aO );
    aU  = wmma4(al, PKu0 [kh], aU );  aU  = wmma4(ar, PKu1 [kh], aU );
  }

  const int  cc  = min(col, HID - 1);
  const bool pad = (col >= HID);
  const float bic = bbi[cc], bfc = bbf[cc], boc = bbo[cc], buc = bbu[cc];
#pragma unroll
  for (int r = 0; r < 8; ++r) {
    int rowd = tm * 16 + r + (half << 3);
    int rc   = min(rowd, W - 1);
    int nd   = s + rc;
    long lid = lchs[nd], rid = rchs[nd];
    float lc = C[(size_t)lid * HP + col];   // padded cols read zeros
    float rcv = C[(size_t)rid * HP + col];
    float ig = sigf(aI [r] + bic);
    float fl = sigf(aFl[r] + bfc);
    float fr = sigf(aFr[r] + bfc);
    float og = sigf(aO [r] + boc);
    float uv = tanhf(aU[r] + buc);
    float c  = ig * uv + fl * lc + fr * rcv;
    float h  = og * tanhf(c);
    if (rowd < W) {
      size_t off = (size_t)(s + rowd) * HP + col;
      H[off] = pad ? 0.0f : h;
      C[off] = pad ? 0.0f : c;
    }
  }
}

// ---------------- loss head ----------------
__global__ void zero_out(float* out) {
  if (blockIdx.x == 0 && threadIdx.x == 0) out[0] = 0.0f;
}

__global__ void loss_kernel(const float* __restrict__ H, const float* __restrict__ Why,
                            const float* __restrict__ by, const int* __restrict__ scores,
                            float* __restrict__ out, int n) {
  int i = blockIdx.x * blockDim.x + threadIdx.x;
  if (i >= n) return;
  const float* h = H + (size_t)i * HP;
  float lg[5];
#pragma unroll
  for (int j = 0; j < 5; ++j) {
    float s = by[j];
    const float* w = Why + j * HID;
    for (int k = 0; k < HID; ++k) s += h[k] * w[k];
    lg[j] = s;
  }
  float m = lg[0];
#pragma unroll
  for (int j = 1; j < 5; ++j) m = fmaxf(m, lg[j]);
  float se = 0.0f;
#pragma unroll
  for (int j = 0; j < 5; ++j) se += expf(lg[j] - m);
  float lse = m + logf(se);
  float loss = lse - lg[scores[i]];
  atomicAdd(out, loss);
}

// ---------------- host side ----------------
extern "C" void kernel_launch(void* const* d_in, const int* in_sizes, int n_in,
                              void* d_out, int out_size, void* d_ws, size_t ws_size,
                              hipStream_t stream) {
  (void)in_sizes; (void)n_in; (void)out_size; (void)ws_size;
  const float* Wi   = (const float*)d_in[0];
  const float* bi   = (const float*)d_in[1];
  const float* Wo   = (const float*)d_in[2];
  const float* bo   = (const float*)d_in[3];
  const float* Wu   = (const float*)d_in[4];
  const float* bu   = (const float*)d_in[5];
  const float* U0i  = (const float*)d_in[6];
  const float* U1i  = (const float*)d_in[7];
  const float* bbi  = (const float*)d_in[8];
  const float* U00f = (const float*)d_in[9];
  const float* U01f = (const float*)d_in[10];
  const float* U10f = (const float*)d_in[11];
  const float* U11f = (const float*)d_in[12];
  const float* bbf  = (const float*)d_in[13];
  const float* U0o  = (const float*)d_in[14];
  const float* U1o  = (const float*)d_in[15];
  const float* bbo  = (const float*)d_in[16];
  const float* U0u  = (const float*)d_in[17];
  const float* U1u  = (const float*)d_in[18];
  const float* bbu  = (const float*)d_in[19];
  const float* Why  = (const float*)d_in[20];
  const float* by   = (const float*)d_in[21];
  const float* emb  = (const float*)d_in[22];
  const int*  scores = (const int*)d_in[23];
  const int*  words  = (const int*)d_in[24];
  const int*  lchs   = (const int*)d_in[25];
  const int*  rchs   = (const int*)d_in[26];

  // workspace layout (floats)
  float* ws = (float*)d_ws;
  float* H  = ws;                                   // 8192*160
  float* C  = H + (size_t)NPADR * HP;               // 8192*160
  float* base = C + (size_t)NPADR * HP;
  const size_t LEAF_PK_F = (size_t)(EMB / 2) * HP * 2;   // floats per leaf PK (150 pairs x 160 v2f)
  const size_t INT_PK_F  = (size_t)(KPAD / 2) * HP * 2;  // floats per internal PK (80 pairs x 160 v2f)
  v2f* WiPK = (v2f*)(base);
  v2f* WoPK = (v2f*)(base + LEAF_PK_F);
  v2f* WuPK = (v2f*)(base + 2 * LEAF_PK_F);
  float* ub = base + 3 * LEAF_PK_F;
  v2f* PK[10];
  for (int t = 0; t < 10; ++t) PK[t] = (v2f*)(ub + (size_t)t * INT_PK_F);
  const float* Usrc[10] = {U0i, U1i, U00f, U01f, U10f, U11f, U0o, U1o, U0u, U1u};

  // pack weights (B-ready, zero padded)
  {
    int totL = (EMB / 2) * HP;                       // 24000
    pack_weight<<<(totL + 255) / 256, 256, 0, stream>>>(Wi, WiPK, HID, EMB, EMB / 2, HP);
    pack_weight<<<(totL + 255) / 256, 256, 0, stream>>>(Wo, WoPK, HID, EMB, EMB / 2, HP);
    pack_weight<<<(totL + 255) / 256, 256, 0, stream>>>(Wu, WuPK, HID, EMB, EMB / 2, HP);
    int totU = (KPAD / 2) * HP;                      // 12800
    for (int t = 0; t < 10; ++t)
      pack_weight<<<(totU + 255) / 256, 256, 0, stream>>>(Usrc[t], PK[t], HID, HID, KPAD / 2, HP);
  }

  zero_out<<<1, 32, 0, stream>>>((float*)d_out);

  // leaves: 4096 rows x 160 cols -> 256 x 10 tiles, one wave each
  {
    dim3 grid(NLEAF / 16, HP / 16);
    leaf_kernel<<<grid, 32, 0, stream>>>(emb, words, WiPK, WoPK, WuPK, bi, bo, bu, H, C);
  }

  // internal levels, root-ward (sequential dependency on stream order)
  for (int d = 11; d >= 0; --d) {
    int s = (1 << d) - 1;
    int W = 1 << d;
    dim3 grid((W + 15) / 16, HP / 16);
    internal_kernel<<<grid, 32, 0, stream>>>(lchs, rchs,
        PK[0], PK[1], PK[2], PK[3], PK[4], PK[5], PK[6], PK[7], PK[8], PK[9],
        bbi, bbf, bbo, bbu, H, C, s, W);
  }

  // loss head
  loss_kernel<<<(NNODE + 255) / 256, 256, 0, stream>>>(H, Why, by, scores, (float*)d_out, NNODE);
}